// AttentionAutoencoder3D_69879117906014
// MI455X (gfx1250) — compile-verified
//
#include <hip/hip_runtime.h>
#include <math.h>

typedef __attribute__((ext_vector_type(2))) float v2f;
typedef __attribute__((ext_vector_type(8))) float v8f;

// =====================================================================
// Layout / shuffle kernels
// =====================================================================

// input x: [8,32,32,32] channel-major -> out: [4096, 64] channels-last (L=16)
__global__ void k_unshuffle_in(const float* __restrict__ x, float* __restrict__ out)
{
    int idx = blockIdx.x * blockDim.x + threadIdx.x;
    if (idx >= 4096 * 64) return;
    int cp = idx & 63;
    int sp = idx >> 6;
    int wp = sp & 15, hp = (sp >> 4) & 15, dp = sp >> 8;
    int c = cp >> 3, rd = (cp >> 2) & 1, rh = (cp >> 1) & 1, rw = cp & 1;
    int d = dp * 2 + rd, h = hp * 2 + rh, w = wp * 2 + rw;
    out[idx] = x[((c * 32 + d) * 32 + h) * 32 + w];
}

// channels-last [ (2L)^3, Cin ] -> channels-last [ L^3, Cin*8 ]
__global__ void k_unshuffle_cl(const float* __restrict__ in, float* __restrict__ out,
                               int L, int Cin)
{
    int Cout = Cin * 8;
    int total = L * L * L * Cout;
    int idx = blockIdx.x * blockDim.x + threadIdx.x;
    if (idx >= total) return;
    int cp = idx % Cout;
    int sp = idx / Cout;
    int wp = sp % L, hp = (sp / L) % L, dp = sp / (L * L);
    int c = cp >> 3, rd = (cp >> 2) & 1, rh = (cp >> 1) & 1, rw = cp & 1;
    int L2 = L * 2;
    int sin = ((dp * 2 + rd) * L2 + (hp * 2 + rh)) * L2 + (wp * 2 + rw);
    out[idx] = in[(size_t)sin * Cin + c];
}

// channels-last [ Lin^3, Cin ] -> channels-last [ (2*Lin)^3, Cin/8 ]
__global__ void k_shuffle_cl(const float* __restrict__ in, float* __restrict__ out,
                             int Lin, int Cin)
{
    int Cout = Cin >> 3;
    int L2 = Lin * 2;
    int total = L2 * L2 * L2 * Cout;
    int idx = blockIdx.x * blockDim.x + threadIdx.x;
    if (idx >= total) return;
    int co = idx % Cout;
    int so = idx / Cout;
    int w2 = so % L2, h2 = (so / L2) % L2, d2 = so / (L2 * L2);
    int d = d2 >> 1, rd = d2 & 1, h = h2 >> 1, rh = h2 & 1, w = w2 >> 1, rw = w2 & 1;
    int cin = co * 8 + rd * 4 + rh * 2 + rw;
    out[idx] = in[(size_t)((d * Lin + h) * Lin + w) * Cin + cin];
}

// =====================================================================
// Normalization stats
// =====================================================================

// BatchNorm: per channel c over S voxels of t[s, c]. y = x*scale[c] + shift[c]
__global__ void k_bn_stats(const float* __restrict__ t, int S, int C,
                           const float* __restrict__ g, const float* __restrict__ b,
                           float* __restrict__ scale, float* __restrict__ shift)
{
    int c = blockIdx.x;
    __shared__ float s1[256], s2[256];
    float a = 0.f, q = 0.f;
    for (int s = threadIdx.x; s < S; s += 256) {
        float v = t[(size_t)s * C + c];
        a += v; q += v * v;
    }
    s1[threadIdx.x] = a; s2[threadIdx.x] = q;
    __syncthreads();
    for (int st = 128; st > 0; st >>= 1) {
        if (threadIdx.x < st) { s1[threadIdx.x] += s1[threadIdx.x + st]; s2[threadIdx.x] += s2[threadIdx.x + st]; }
        __syncthreads();
    }
    if (threadIdx.x == 0) {
        float m = s1[0] / (float)S;
        float var = s2[0] / (float)S - m * m;
        float sc = g[c] * rsqrtf(var + 1e-5f);
        scale[c] = sc;
        shift[c] = b[c] - m * sc;
    }
}

// LayerNorm row stats: y = (x*rs[row] + rb[row]) * g[c] + b[c]
__global__ void k_ln_stats(const float* __restrict__ t, int C,
                           float* __restrict__ rs, float* __restrict__ rb)
{
    int row = blockIdx.x;
    const float* p = t + (size_t)row * C;
    __shared__ float s1[256], s2[256];
    float a = 0.f, q = 0.f;
    for (int i = threadIdx.x; i < C; i += 256) {
        float v = p[i];
        a += v; q += v * v;
    }
    s1[threadIdx.x] = a; s2[threadIdx.x] = q;
    __syncthreads();
    for (int st = 128; st > 0; st >>= 1) {
        if (threadIdx.x < st) { s1[threadIdx.x] += s1[threadIdx.x + st]; s2[threadIdx.x] += s2[threadIdx.x + st]; }
        __syncthreads();
    }
    if (threadIdx.x == 0) {
        float m = s1[0] / (float)C;
        float var = s2[0] / (float)C - m * m;
        float r = rsqrtf(var + 1e-5f);
        rs[row] = r;
        rb[row] = -m * r;
    }
}

// Fold per-k column affine into weights:
//   Bf[n,k] = B[n,k]*cs[k];  biasf[n] = bias[n] + sum_k B[n,k]*cb[k]
__global__ void k_fold_w(const float* __restrict__ Bw, int K,
                         const float* __restrict__ cs, const float* __restrict__ cb,
                         const float* __restrict__ bias,
                         float* __restrict__ Bf, float* __restrict__ biasf)
{
    int n = blockIdx.x;
    const float* src = Bw + (size_t)n * K;
    float* dst = Bf + (size_t)n * K;
    __shared__ float sm[256];
    float acc = 0.f;
    for (int k = threadIdx.x; k < K; k += 256) {
        float w = src[k];
        dst[k] = w * cs[k];
        acc += w * cb[k];
    }
    sm[threadIdx.x] = acc;
    __syncthreads();
    for (int st = 128; st > 0; st >>= 1) {
        if (threadIdx.x < st) sm[threadIdx.x] += sm[threadIdx.x + st];
        __syncthreads();
    }
    if (threadIdx.x == 0) biasf[n] = bias[n] + sm[0];
}

// row softmax (in place)
__global__ void k_softmax_rows(float* __restrict__ x, int n)
{
    float* p = x + (size_t)blockIdx.x * n;
    __shared__ float sm[256];
    float mx = -3.0e38f;
    for (int i = threadIdx.x; i < n; i += 256) mx = fmaxf(mx, p[i]);
    sm[threadIdx.x] = mx;
    __syncthreads();
    for (int st = 128; st > 0; st >>= 1) {
        if (threadIdx.x < st) sm[threadIdx.x] = fmaxf(sm[threadIdx.x], sm[threadIdx.x + st]);
        __syncthreads();
    }
    mx = sm[0];
    __syncthreads();
    float sum = 0.f;
    for (int i = threadIdx.x; i < n; i += 256) {
        float e = __expf(p[i] - mx);
        p[i] = e;
        sum += e;
    }
    sm[threadIdx.x] = sum;
    __syncthreads();
    for (int st = 128; st > 0; st >>= 1) {
        if (threadIdx.x < st) sm[threadIdx.x] += sm[threadIdx.x + st];
        __syncthreads();
    }
    float inv = 1.f / sm[0];
    for (int i = threadIdx.x; i < n; i += 256) p[i] *= inv;
}

// =====================================================================
// Branch-free register-blocked WMMA fp32 GEMM (32x32 tile per wave32):
//   D = act( alpha * A' x op(B) + bias ) + resid
//   A'[m,k] = A[m,k]*rowS[m] + rowB[m]   (compile-time ROWAFF)
//   op(B)   = B^T (BT: B row-major [N,K]) or B (B is [K,N])
// REQUIRES: Kd % 4 == 0; all lda/ldb even (guaranteed by call sites).
// Tail rows/cols are CLAMPED (garbage accumulates only into unstored
// positions). 2x2 fragment blocking: 4 wmmas per 4 b64 loads, all
// accumulators independent (no WMMA->WMMA RAW hazards).
// =====================================================================
template <bool BT, bool ROWAFF>
__global__ void k_wmma_gemm(const float* __restrict__ A, int lda,
                            const float* __restrict__ B, int ldb,
                            float* __restrict__ D, int ldd,
                            int M, int N, int Kd,
                            const float* __restrict__ bias,
                            const float* __restrict__ rowS, const float* __restrict__ rowB,
                            float alpha,
                            const float* __restrict__ resid, int ldres,
                            int act, int storeT)
{
    const int lane  = threadIdx.x & 31;
    const int half  = lane >> 4;   // selects K pair {0,1} vs {2,3} in each step
    const int r     = lane & 15;
    const int nBase = blockIdx.x * 32;
    const int mBase = blockIdx.y * 32;

    int ma0 = mBase + r;      if (ma0 >= M) ma0 = M - 1;   // clamped addressing
    int ma1 = mBase + 16 + r; if (ma1 >= M) ma1 = M - 1;
    int nb0 = nBase + r;      if (nb0 >= N) nb0 = N - 1;
    int nb1 = nBase + 16 + r; if (nb1 >= N) nb1 = N - 1;

    float rs0 = 1.f, rb0 = 0.f, rs1 = 1.f, rb1 = 0.f;
    if (ROWAFF) {
        rs0 = rowS[ma0]; rb0 = rowB[ma0];
        rs1 = rowS[ma1]; rb1 = rowB[ma1];
    }

    const float* ap0 = A + (size_t)ma0 * lda + 2 * half;
    const float* ap1 = A + (size_t)ma1 * lda + 2 * half;
    const float* bp0 = BT ? (B + (size_t)nb0 * ldb + 2 * half)
                          : (B + (size_t)(2 * half) * ldb + nb0);
    const float* bp1 = BT ? (B + (size_t)nb1 * ldb + 2 * half)
                          : (B + (size_t)(2 * half) * ldb + nb1);
    const size_t bstep = BT ? 4 : (size_t)4 * ldb;

    v8f acc00 = {}, acc01 = {}, acc10 = {}, acc11 = {};
#pragma unroll 2
    for (int k0 = 0; k0 < Kd; k0 += 4) {
        v2f a0 = *(const v2f*)ap0;
        v2f a1 = *(const v2f*)ap1;
        if (ROWAFF) {
            a0[0] = a0[0] * rs0 + rb0; a0[1] = a0[1] * rs0 + rb0;
            a1[0] = a1[0] * rs1 + rb1; a1[1] = a1[1] * rs1 + rb1;
        }
        v2f b0, b1;
        if (BT) {
            b0 = *(const v2f*)bp0;
            b1 = *(const v2f*)bp1;
        } else {
            b0[0] = bp0[0]; b0[1] = bp0[ldb];
            b1[0] = bp1[0]; b1[1] = bp1[ldb];
        }
        acc00 = __builtin_amdgcn_wmma_f32_16x16x4_f32(false, a0, false, b0, (short)0, acc00, false, false);
        acc01 = __builtin_amdgcn_wmma_f32_16x16x4_f32(false, a0, false, b1, (short)0, acc01, false, false);
        acc10 = __builtin_amdgcn_wmma_f32_16x16x4_f32(false, a1, false, b0, (short)0, acc10, false, false);
        acc11 = __builtin_amdgcn_wmma_f32_16x16x4_f32(false, a1, false, b1, (short)0, acc11, false, false);
        ap0 += 4; ap1 += 4;
        bp0 += bstep; bp1 += bstep;
    }

    // Epilogue: per fragment, VGPR i: lanes 0-15 -> M=i, lanes 16-31 -> M=i+8
    const int col0 = nBase + r;
    const int col1 = nBase + 16 + r;
#pragma unroll
    for (int i = 0; i < 8; ++i) {
        int row0 = mBase + i + 8 * half;
        int row1 = row0 + 16;
        float vals[4] = { acc00[i], acc01[i], acc10[i], acc11[i] };
        int   rows[4] = { row0, row0, row1, row1 };
        int   cols[4] = { col0, col1, col0, col1 };
#pragma unroll
        for (int j = 0; j < 4; ++j) {
            int row = rows[j], col = cols[j];
            if (row < M && col < N) {
                float v = vals[j] * alpha;
                if (bias) v += bias[col];
                if (act == 1)  // exact GELU
                    v = 0.5f * v * (1.0f + erff(v * 0.70710678118654752440f));
                if (resid) v += resid[(size_t)row * ldres + col];
                if (storeT) D[(size_t)col * ldd + row] = v;
                else        D[(size_t)row * ldd + col] = v;
            }
        }
    }
}

// =====================================================================
// 3D neighborhood attention (K=5, 125 taps), online softmax.
// One thread per (voxel, head); HD templated so q/acc stay in registers.
// qkv: [S, 3C] (q | k | v), rpb: [8, 9, 9, 9], out: [S, C]
// =====================================================================
template <int HD>
__global__ void k_na_attn(const float* __restrict__ qkv, const float* __restrict__ rpb,
                          float* __restrict__ out, int L, int C)
{
    int tid = blockIdx.x * blockDim.x + threadIdx.x;
    int S = L * L * L;
    if (tid >= S * 8) return;
    int n = tid & 7;
    int s = tid >> 3;
    int w = s % L, h = (s / L) % L, d = s / (L * L);
    int sd = d - 2; sd = sd < 0 ? 0 : (sd > L - 5 ? L - 5 : sd);
    int sh = h - 2; sh = sh < 0 ? 0 : (sh > L - 5 ? L - 5 : sh);
    int sw = w - 2; sw = sw < 0 ? 0 : (sw > L - 5 ? L - 5 : sw);

    const float scale = rsqrtf((float)HD);
    const float* qrow = qkv + (size_t)s * 3 * C + n * HD;
    float q[HD], acc[HD];
    for (int e = 0; e < HD; ++e) { q[e] = qrow[e] * scale; acc[e] = 0.f; }

    float m = -3.0e38f, l = 0.f;
    for (int a = 0; a < 5; ++a)
        for (int p = 0; p < 5; ++p)
            for (int qq = 0; qq < 5; ++qq) {
                int nd = sd + a, nh = sh + p, nw = sw + qq;
                int ns = (nd * L + nh) * L + nw;
                const float* kr = qkv + (size_t)ns * 3 * C + C + n * HD;
                float lg = 0.f;
                for (int e = 0; e < HD; ++e) lg += q[e] * kr[e];
                lg += rpb[((n * 9 + (nd - d + 4)) * 9 + (nh - h + 4)) * 9 + (nw - w + 4)];
                float mn = fmaxf(m, lg);
                float corr = __expf(m - mn);
                float pe   = __expf(lg - mn);
                l = l * corr + pe;
                const float* vr = qkv + (size_t)ns * 3 * C + 2 * C + n * HD;
                for (int e = 0; e < HD; ++e) acc[e] = acc[e] * corr + pe * vr[e];
                m = mn;
            }
    float inv = 1.f / l;
    float* orow = out + (size_t)s * C + n * HD;
    for (int e = 0; e < HD; ++e) orow[e] = acc[e] * inv;
}

// =====================================================================
// Host-side orchestration
// =====================================================================

static void launch_gemm(bool BT, bool rowaff,
                        const float* A, int lda, const float* B, int ldb,
                        float* D, int ldd, int M, int N, int K,
                        const float* bias,
                        const float* rowS, const float* rowB,
                        float alpha, const float* resid, int ldres,
                        int act, int storeT, hipStream_t stream)
{
    dim3 grid((N + 31) / 32, (M + 31) / 32);
    dim3 blk(32);
    if (BT) {
        if (rowaff)
            k_wmma_gemm<true, true><<<grid, blk, 0, stream>>>(A, lda, B, ldb, D, ldd, M, N, K,
                                                              bias, rowS, rowB, alpha, resid, ldres, act, storeT);
        else
            k_wmma_gemm<true, false><<<grid, blk, 0, stream>>>(A, lda, B, ldb, D, ldd, M, N, K,
                                                               bias, rowS, rowB, alpha, resid, ldres, act, storeT);
    } else {
        if (rowaff)
            k_wmma_gemm<false, true><<<grid, blk, 0, stream>>>(A, lda, B, ldb, D, ldd, M, N, K,
                                                               bias, rowS, rowB, alpha, resid, ldres, act, storeT);
        else
            k_wmma_gemm<false, false><<<grid, blk, 0, stream>>>(A, lda, B, ldb, D, ldd, M, N, K,
                                                                bias, rowS, rowB, alpha, resid, ldres, act, storeT);
    }
}

struct NAP {
    const float *bn1_g, *bn1_b, *qkv_w, *qkv_b, *rpb, *proj_w, *proj_b,
                *bn2_g, *bn2_b, *mlp_w1, *mlp_b1, *mlp_w2, *mlp_b2;
};
struct GABP {
    const float *ln1_g, *ln1_b, *in_w, *in_b, *out_w, *out_b,
                *ln2_g, *ln2_b, *mlp_w1, *mlp_b1, *mlp_w2, *mlp_b2;
};

// NA block, channels-last carrier t [S, C]; result left in t. t2 is scratch.
static void na_block(float* t, float* t2, float* QKV, float* ATT, float* HID,
                     float* ST1, float* ST2, float* WF, float* BF,
                     int L, int C, const NAP& p, hipStream_t stream)
{
    const int S = L * L * L;
    const int hd = C / 8;
    // BN1 folded into qkv weights
    k_bn_stats<<<C, 256, 0, stream>>>(t, S, C, p.bn1_g, p.bn1_b, ST1, ST2);
    k_fold_w<<<3 * C, 256, 0, stream>>>(p.qkv_w, C, ST1, ST2, p.qkv_b, WF, BF);
    launch_gemm(true, false, t, C, WF, C, QKV, 3 * C, S, 3 * C, C, BF,
                nullptr, nullptr, 1.f, nullptr, 0, 0, 0, stream);
    // neighborhood attention
    {
        int tot = S * 8, thr = 256, blk = (tot + thr - 1) / thr;
        if (hd == 1)       k_na_attn<1><<<blk, thr, 0, stream>>>(QKV, p.rpb, ATT, L, C);
        else if (hd == 8)  k_na_attn<8><<<blk, thr, 0, stream>>>(QKV, p.rpb, ATT, L, C);
        else               k_na_attn<64><<<blk, thr, 0, stream>>>(QKV, p.rpb, ATT, L, C);
    }
    // proj + residual
    launch_gemm(true, false, ATT, C, p.proj_w, C, t2, C, S, C, C, p.proj_b,
                nullptr, nullptr, 1.f, t, C, 0, 0, stream);
    // BN2 folded into mlp1 weights; GELU fused
    k_bn_stats<<<C, 256, 0, stream>>>(t2, S, C, p.bn2_g, p.bn2_b, ST1, ST2);
    k_fold_w<<<4 * C, 256, 0, stream>>>(p.mlp_w1, C, ST1, ST2, p.mlp_b1, WF, BF);
    launch_gemm(true, false, t2, C, WF, C, HID, 4 * C, S, 4 * C, C, BF,
                nullptr, nullptr, 1.f, nullptr, 0, 1, 0, stream);
    launch_gemm(true, false, HID, 4 * C, p.mlp_w2, 4 * C, t, C, S, C, 4 * C, p.mlp_b2,
                nullptr, nullptr, 1.f, t2, C, 0, 0, stream);
}

// Global attention block on tokens t [512, 512]; result left in t.
static void gab_block(float* t, float* t2, float* QKV, float* ATT, float* HID,
                      float* ST1, float* ST2, float* WF, float* BF,
                      const GABP& p, hipStream_t stream)
{
    const int N = 512, C = 512, hd = 64;
    // LN1: gamma/beta folded into in_w; per-row part via ROWAFF
    k_ln_stats<<<N, 256, 0, stream>>>(t, C, ST1, ST2);
    k_fold_w<<<3 * C, 256, 0, stream>>>(p.in_w, C, p.ln1_g, p.ln1_b, p.in_b, WF, BF);
    launch_gemm(true, true, t, C, WF, C, QKV, 3 * C, N, 3 * C, C, BF,
                ST1, ST2, 1.f, nullptr, 0, 0, 0, stream);
    // per-head logits = alpha * Q K^T
    for (int h = 0; h < 8; ++h)
        launch_gemm(true, false, QKV + h * hd, 3 * C, QKV + C + h * hd, 3 * C,
                    HID + (size_t)h * N * N, N, N, N, hd,
                    nullptr, nullptr, nullptr, 0.125f, nullptr, 0, 0, 0, stream);
    k_softmax_rows<<<8 * N, 256, 0, stream>>>(HID, N);
    // O_h = A V_h  (B not transposed: V is [K rows, hd cols] strided in QKV)
    for (int h = 0; h < 8; ++h)
        launch_gemm(false, false, HID + (size_t)h * N * N, N, QKV + 2 * C + h * hd, 3 * C,
                    ATT + h * hd, C, N, hd, N,
                    nullptr, nullptr, nullptr, 1.f, nullptr, 0, 0, 0, stream);
    // out proj + residual
    launch_gemm(true, false, ATT, C, p.out_w, C, t2, C, N, C, C, p.out_b,
                nullptr, nullptr, 1.f, t, C, 0, 0, stream);
    // LN2 + MLP (GELU fused) + residual
    k_ln_stats<<<N, 256, 0, stream>>>(t2, C, ST1, ST2);
    k_fold_w<<<4 * C, 256, 0, stream>>>(p.mlp_w1, C, p.ln2_g, p.ln2_b, p.mlp_b1, WF, BF);
    launch_gemm(true, true, t2, C, WF, C, HID, 4 * C, N, 4 * C, C, BF,
                ST1, ST2, 1.f, nullptr, 0, 1, 0, stream);
    launch_gemm(true, false, HID, 4 * C, p.mlp_w2, 4 * C, t, C, N, C, 4 * C, p.mlp_b2,
                nullptr, nullptr, 1.f, t2, C, 0, 0, stream);
}

extern "C" void kernel_launch(void* const* d_in, const int* in_sizes, int n_in,
                              void* d_out, int out_size, void* d_ws, size_t ws_size,
                              hipStream_t stream)
{
    (void)in_sizes; (void)n_in; (void)out_size; (void)ws_size;

    // ---- unpack parameters (setup_inputs dict insertion order) ----
    int i = 0;
    const float* x = (const float*)d_in[i++];
    auto nextNA = [&](NAP& p) {
        p.bn1_g  = (const float*)d_in[i++]; p.bn1_b  = (const float*)d_in[i++];
        p.qkv_w  = (const float*)d_in[i++]; p.qkv_b  = (const float*)d_in[i++];
        p.rpb    = (const float*)d_in[i++];
        p.proj_w = (const float*)d_in[i++]; p.proj_b = (const float*)d_in[i++];
        p.bn2_g  = (const float*)d_in[i++]; p.bn2_b  = (const float*)d_in[i++];
        p.mlp_w1 = (const float*)d_in[i++]; p.mlp_b1 = (const float*)d_in[i++];
        p.mlp_w2 = (const float*)d_in[i++]; p.mlp_b2 = (const float*)d_in[i++];
    };
    auto nextGAB = [&](GABP& p) {
        p.ln1_g  = (const float*)d_in[i++]; p.ln1_b  = (const float*)d_in[i++];
        p.in_w   = (const float*)d_in[i++]; p.in_b   = (const float*)d_in[i++];
        p.out_w  = (const float*)d_in[i++]; p.out_b  = (const float*)d_in[i++];
        p.ln2_g  = (const float*)d_in[i++]; p.ln2_b  = (const float*)d_in[i++];
        p.mlp_w1 = (const float*)d_in[i++]; p.mlp_b1 = (const float*)d_in[i++];
        p.mlp_w2 = (const float*)d_in[i++]; p.mlp_b2 = (const float*)d_in[i++];
    };
    NAP enc1; nextNA(enc1);
    NAP enc2; nextNA(enc2);
    GABP gb0; nextGAB(gb0);
    GABP gb1; nextGAB(gb1);
    NAP dec2; nextNA(dec2);
    NAP dec1; nextNA(dec1);
    const float* out_w = (const float*)d_in[i++];
    const float* out_b = (const float*)d_in[i++];

    // ---- workspace carve (floats); everything stays L2-resident (~19 MB) ----
    float* W = (float*)d_ws;
    size_t off = 0;
    auto carve = [&](size_t n) { float* p = W + off; off += n; return p; };
    float* tA  = carve(262144);   // carrier (max [32768,8]/[4096,64]/[512,512])
    float* tB  = carve(262144);   // second carrier / block scratch
    float* QKV = carve(786432);   // [S, 3C] max
    float* ATT = carve(262144);   // [S, C] max
    float* HID = carve(2097152);  // MLP hidden (max 1M) / GAB logits (2M)
    float* ST1 = carve(4096);     // scale / row rsqrt
    float* ST2 = carve(4096);     // shift / row bias
    float* WF  = carve(1048576);  // norm-folded weights (max 2048x512)
    float* BF  = carve(2048);     // norm-folded bias

    const int THR = 256;

    // ---- encoder ----
    k_unshuffle_in<<<(4096 * 64 + THR - 1) / THR, THR, 0, stream>>>(x, tA);
    na_block(tA, tB, QKV, ATT, HID, ST1, ST2, WF, BF, 16, 64, enc1, stream);

    k_unshuffle_cl<<<(512 * 512 + THR - 1) / THR, THR, 0, stream>>>(tA, tB, 8, 64);
    na_block(tB, tA, QKV, ATT, HID, ST1, ST2, WF, BF, 8, 512, enc2, stream);

    // ---- bottleneck (tokens [512, 512] == channels-last carrier) ----
    gab_block(tB, tA, QKV, ATT, HID, ST1, ST2, WF, BF, gb0, stream);
    gab_block(tB, tA, QKV, ATT, HID, ST1, ST2, WF, BF, gb1, stream);

    // ---- decoder ----
    k_shuffle_cl<<<(4096 * 64 + THR - 1) / THR, THR, 0, stream>>>(tB, tA, 8, 512);
    na_block(tA, tB, QKV, ATT, HID, ST1, ST2, WF, BF, 16, 64, dec2, stream);

    k_shuffle_cl<<<(32768 * 8 + THR - 1) / THR, THR, 0, stream>>>(tA, tB, 16, 64);
    na_block(tB, tA, QKV, ATT, HID, ST1, ST2, WF, BF, 32, 8, dec1, stream);

    // ---- final 1x1 conv, transposed store -> [8, 32, 32, 32] channel-major ----
    launch_gemm(true, false, tB, 8, out_w, 8, (float*)d_out, 32768, 32768, 8, 8, out_b,
                nullptr, nullptr, 1.f, nullptr, 0, 0, 1, stream);
}